// BinsChamferLoss_16200616640818
// MI455X (gfx1250) — compile-verified
//
#include <hip/hip_runtime.h>
#include <hip/hip_bf16.h>
#include <stdint.h>

// Problem constants (from reference setup_inputs)
#define L_SCALES 4
#define NBATCH   4
#define P1       257
#define PBINS    256
#define MPTS     (240 * 320)      // 76800 points per batch
#define EPS_DEPTH 0.001f
#define BIG_F     1e10f
#define YTILES   (MPTS / 16)      // 4800 tiles of 16 points

#define BLOCKS_PER_LN   60
#define WAVES_PER_BLOCK 8
#define THREADS         (WAVES_PER_BLOCK * 32)
#define TOTAL_WAVES_LN  (BLOCKS_PER_LN * WAVES_PER_BLOCK)  // 480 waves per (l,n)
// 4800 tiles / 480 waves = 10 tiles per wave -> 5 double-tile iterations

#define INF_BITS 0x7F800000

typedef __attribute__((ext_vector_type(2))) float v2f;
typedef __attribute__((ext_vector_type(8))) float v8f;

// Signed-int min == float min for non-negative floats; tiny-negative bit
// patterns (from c^2-2cy+y^2 cancellation) still sort below all positives,
// which is the correct direction for a min. Avoids v_max canonicalize spam.
__device__ __forceinline__ int imin2(int a, int b) { return a < b ? a : b; }
__device__ __forceinline__ int imin3(int a, int b, int c) {
    return imin2(imin2(a, b), c);   // -> v_min3_i32
}

// ---------------------------------------------------------------------------
// Kernel 1: initialize workspace (minx = +inf bits, sums/counts = 0)
// ---------------------------------------------------------------------------
__global__ void init_ws_kernel(int* __restrict__ minxBits,
                               float* __restrict__ sumy,
                               float* __restrict__ cnt) {
    int i = blockIdx.x * blockDim.x + threadIdx.x;
    if (i < L_SCALES * NBATCH * PBINS) minxBits[i] = INF_BITS;
    if (i < L_SCALES * NBATCH)         sumy[i]     = 0.0f;
    if (i < NBATCH)                    cnt[i]      = 0.0f;
}

// ---------------------------------------------------------------------------
// Kernel 2: count valid depth points per batch (wave-aggregated)
// ---------------------------------------------------------------------------
__global__ void count_valid_kernel(const float* __restrict__ y,
                                   float* __restrict__ cnt) {
    int i = blockIdx.x * blockDim.x + threadIdx.x;   // [0, NBATCH*MPTS)
    int n = i / MPTS;                                 // uniform per wave (MPTS % 32 == 0)
    bool valid = y[i] >= EPS_DEPTH;
    unsigned long long m = __ballot(valid);
    if ((threadIdx.x & 31) == 0)
        atomicAdd(&cnt[n], (float)__popcll(m));
}

// ---------------------------------------------------------------------------
// Kernel 3: WMMA chamfer. One (l,n) pair per BLOCKS_PER_LN blocks.
//
// Distance trick: (c-y)^2 = [c^2, -2c, 1, 0] . [1, y, y^2, 0]
// so V_WMMA_F32_16X16X4_F32 produces a 16(centers) x 16(points) tile of
// squared distances per instruction. Two y-tiles per rolled iteration so the
// per-center accumulator update is a single v_min3_i32.
// __launch_bounds__(.., 4): cap at 1024/4 = 256 VGPRs, avoiding spills and
// VGPR-MSB addressing (seen when fully unrolled).
// ---------------------------------------------------------------------------
__global__ void __launch_bounds__(THREADS, 4)
chamfer_wmma_kernel(const float* __restrict__ bins,   // [L, N, P1]
                    const float* __restrict__ y,      // [N, MPTS]
                    int* __restrict__ minxBits,       // [L*N, PBINS]
                    float* __restrict__ sumy) {       // [L*N]
    const int ln      = blockIdx.x / BLOCKS_PER_LN;   // l*NBATCH + n
    const int chunk   = blockIdx.x % BLOCKS_PER_LN;
    const int n       = ln % NBATCH;
    const int lane    = threadIdx.x & 31;
    const int wave    = threadIdx.x >> 5;
    const int row     = lane & 15;
    const bool hiHalf = (lane >= 16);

    // ---- Build 16 A fragments covering all 256 centers of this (l,n) ----
    const float* binRow = bins + (size_t)ln * P1;
    v2f aF[16];
#pragma unroll
    for (int t = 0; t < 16; ++t) {
        int p   = t * 16 + row;
        float c = 0.5f * (binRow[p] + binRow[p + 1]);
        v2f a;
        a.x = hiHalf ? 1.0f : c * c;      // K=0 (lo) / K=2 (hi)
        a.y = hiHalf ? 0.0f : -2.0f * c;  // K=1 (lo) / K=3 (hi)
        aF[t] = a;
    }

    // Running per-center min over y (int-domain, elementwise across y-tiles)
    int accX[16][8];
#pragma unroll
    for (int t = 0; t < 16; ++t)
#pragma unroll
        for (int r = 0; r < 8; ++r) accX[t][r] = INF_BITS;

    const float* yb = y + (size_t)n * MPTS;
    float sumAcc = 0.0f;

    // Rolled loop: tiles (tile, tile+480); 5 iterations per wave.
#pragma clang loop unroll(disable)
    for (int tile = chunk * WAVES_PER_BLOCK + wave; tile < YTILES;
         tile += 2 * TOTAL_WAVES_LN) {
        // Issue both loads back-to-back so they overlap.
        float y0 = yb[tile * 16 + row];
        float y1 = yb[(tile + TOTAL_WAVES_LN) * 16 + row];

        bool  v0 = (y0 >= EPS_DEPTH);
        bool  v1 = (y1 >= EPS_DEPTH);
        float s0 = v0 ? y0 : BIG_F;   // invalid -> never wins center-side min
        float s1 = v1 ? y1 : BIG_F;

        v2f b0, b1;
        b0.x = hiHalf ? s0 * s0 : 1.0f;   // K=0 / K=2
        b0.y = hiHalf ? 0.0f    : s0;     // K=1 / K=3
        b1.x = hiHalf ? s1 * s1 : 1.0f;
        b1.y = hiHalf ? 0.0f    : s1;

        int minY0[8], minY1[8];

        // ---- tile t = 0 peeled: initializes minY (no INF mins) ----
        {
            v8f cz0 = {}, cz1 = {};
            v8f d0 = __builtin_amdgcn_wmma_f32_16x16x4_f32(
                false, aF[0], false, b0, (short)0, cz0, false, false);
            v8f d1 = __builtin_amdgcn_wmma_f32_16x16x4_f32(
                false, aF[0], false, b1, (short)0, cz1, false, false);
#pragma unroll
            for (int r = 0; r < 8; ++r) {
                int i0 = __float_as_int(d0[r]);
                int i1 = __float_as_int(d1[r]);
                accX[0][r] = imin3(accX[0][r], i0, i1);
                minY0[r] = i0;
                minY1[r] = i1;
            }
        }
#pragma unroll
        for (int t = 1; t < 16; ++t) {
            v8f cz0 = {}, cz1 = {};
            v8f d0 = __builtin_amdgcn_wmma_f32_16x16x4_f32(
                false, aF[t], false, b0, (short)0, cz0, false, false);
            v8f d1 = __builtin_amdgcn_wmma_f32_16x16x4_f32(
                false, aF[t], false, b1, (short)0, cz1, false, false);
#pragma unroll
            for (int r = 0; r < 8; ++r) {
                int i0 = __float_as_int(d0[r]);
                int i1 = __float_as_int(d1[r]);
                accX[t][r] = imin3(accX[t][r], i0, i1);  // one v_min3_i32
                minY0[r]   = imin2(minY0[r], i0);        // reassoc -> min3 trees
                minY1[r]   = imin2(minY1[r], i1);
            }
        }

        // min over centers for each point: 8 regs, then combine lane halves
        int m0 = imin2(imin3(imin3(minY0[0], minY0[1], minY0[2]),
                             minY0[3], minY0[4]),
                       imin3(minY0[5], minY0[6], minY0[7]));
        int m1 = imin2(imin3(imin3(minY1[0], minY1[1], minY1[2]),
                             minY1[3], minY1[4]),
                       imin3(minY1[5], minY1[6], minY1[7]));
        m0 = imin2(m0, __shfl_xor(m0, 16, 32));   // rows 0-7 vs rows 8-15
        m1 = imin2(m1, __shfl_xor(m1, 16, 32));
        // gate via select (keeps EXEC intact around WMMAs)
        sumAcc += (v0 && !hiHalf) ? __int_as_float(m0) : 0.0f;
        sumAcc += (v1 && !hiHalf) ? __int_as_float(m1) : 0.0f;
    }

    // ---- Reduce per-center mins across the 16 point-columns, commit ----
    int* mBase = minxBits + (size_t)ln * PBINS;
#pragma unroll
    for (int t = 0; t < 16; ++t) {
#pragma unroll
        for (int r = 0; r < 8; ++r) {
            int v = accX[t][r];
            v = imin2(v, __shfl_xor(v, 1, 32));
            v = imin2(v, __shfl_xor(v, 2, 32));
            v = imin2(v, __shfl_xor(v, 4, 32));
            v = imin2(v, __shfl_xor(v, 8, 32));
            if (row == 0) {  // lanes 0 and 16 hold the two half-mins
                int p = t * 16 + r + (hiHalf ? 8 : 0);
                atomicMin(&mBase[p], v);  // signed-int min == float min here
            }
        }
    }

    // ---- Reduce point-side sum across the wave, one atomic per wave ----
    sumAcc += __shfl_xor(sumAcc, 1, 32);
    sumAcc += __shfl_xor(sumAcc, 2, 32);
    sumAcc += __shfl_xor(sumAcc, 4, 32);
    sumAcc += __shfl_xor(sumAcc, 8, 32);
    sumAcc += __shfl_xor(sumAcc, 16, 32);
    if (lane == 0) atomicAdd(&sumy[ln], sumAcc);
}

// ---------------------------------------------------------------------------
// Kernel 4: finalize scalar loss
// loss = sum_{l,n,p} clamp(minx, 0, BIG)/(P*N) + sum_{l,n} sumy/(cnt[n]*N)
// ---------------------------------------------------------------------------
__global__ void finalize_kernel(const int* __restrict__ minxBits,
                                const float* __restrict__ sumy,
                                const float* __restrict__ cnt,
                                float* __restrict__ out) {
    __shared__ float red[256];
    int t = threadIdx.x;
    float acc = 0.0f;
    for (int i = t; i < L_SCALES * NBATCH * PBINS; i += 256) {
        float v = __int_as_float(minxBits[i]);
        v = fmaxf(v, 0.0f);          // undo any cancellation negatives
        acc += fminf(v, BIG_F);      // all-invalid batch -> BIG, as in reference
    }
    acc *= 1.0f / (float)(PBINS * NBATCH);
    if (t < L_SCALES * NBATCH) {
        int n = t % NBATCH;
        acc += sumy[t] / cnt[n] / (float)NBATCH;
    }
    red[t] = acc;
    __syncthreads();
    for (int s = 128; s > 0; s >>= 1) {
        if (t < s) red[t] += red[t + s];
        __syncthreads();
    }
    if (t == 0) out[0] = red[0];
}

// ---------------------------------------------------------------------------
extern "C" void kernel_launch(void* const* d_in, const int* in_sizes, int n_in,
                              void* d_out, int out_size, void* d_ws, size_t ws_size,
                              hipStream_t stream) {
    const float* bins = (const float*)d_in[0];   // [L, N, P1] f32
    const float* y    = (const float*)d_in[1];   // [N, H, W]  f32
    float* out        = (float*)d_out;           // scalar f32

    uint8_t* ws = (uint8_t*)d_ws;
    int* minxBits = (int*)ws;                                          // 4096 i32
    float* sumy = (float*)(ws + (size_t)L_SCALES * NBATCH * PBINS * 4); // 16 f32
    float* cnt  = sumy + L_SCALES * NBATCH;                            // 4 f32

    init_ws_kernel<<<(L_SCALES * NBATCH * PBINS + 255) / 256, 256, 0, stream>>>(
        minxBits, sumy, cnt);
    count_valid_kernel<<<(NBATCH * MPTS) / 256, 256, 0, stream>>>(y, cnt);
    chamfer_wmma_kernel<<<L_SCALES * NBATCH * BLOCKS_PER_LN, THREADS, 0, stream>>>(
        bins, y, minxBits, sumy);
    finalize_kernel<<<1, 256, 0, stream>>>(minxBits, sumy, cnt, out);
}